// ConvCaps_27487790695236
// MI455X (gfx1250) — compile-verified
//
#include <hip/hip_runtime.h>
#include <hip/hip_bf16.h>
#include <stdint.h>

namespace {
constexpr int Bz   = 4;
constexpr int H    = 16;
constexpr int W    = 16;
constexpr int OHh  = 14;
constexpr int OWw  = 14;
constexpr int IC   = 32;
constexpr int OC   = 32;
constexpr int ATOM = 16;                    // 4x4 pose atoms
constexpr int SLAB = IC * OC * ATOM;        // 16384 floats per (ky,kx) = 64 KB
constexpr int NKK  = 9;                     // 3x3 kernel taps
constexpr int PIXV = NKK * SLAB;            // 147456 vote floats per output pixel
constexpr int NPIX = Bz * OHh * OWw;        // 784 output pixels
constexpr long VOTES_TOTAL = (long)NPIX * PIXV;  // 115,605,504 floats
constexpr int ROWF = 3 * IC * ATOM;         // 1536 floats: one pose row, kx=0..2
constexpr int THREADS = 256;                // 8 wave32 per block
}

typedef float f4  __attribute__((ext_vector_type(4)));
typedef int   v4i __attribute__((ext_vector_type(4)));

// Address-space-qualified int4 pointer types matching the builtin's signature
// (clang diagnostic showed arg0 as 'int ext_vector(4) __device__ *', i.e. AS1).
typedef __attribute__((address_space(1))) v4i  as1_v4i;
typedef __attribute__((address_space(3))) v4i  as3_v4i;

#if __has_builtin(__builtin_amdgcn_global_load_async_to_lds_b128)
#define HAVE_ASYNC_LDS 1
#else
#define HAVE_ASYNC_LDS 0
#endif

// 16-byte global -> LDS copy. Async (ASYNCcnt) path on gfx1250 when the
// builtin exists; falls back to a synchronous register bounce otherwise.
__device__ __forceinline__ void cp16_g2l(const float* g, float* l) {
#if HAVE_ASYNC_LDS
  __builtin_amdgcn_global_load_async_to_lds_b128(
      (as1_v4i*)g, (as3_v4i*)l, /*imm offset*/ 0, /*cpol*/ 0);
#else
  *(f4*)l = *(const f4*)g;
#endif
}

__device__ __forceinline__ void async_wait_all() {
#if HAVE_ASYNC_LDS
#if __has_builtin(__builtin_amdgcn_s_wait_asynccnt)
  __builtin_amdgcn_s_wait_asynccnt(0);
#else
  asm volatile("s_wait_asynccnt 0" ::: "memory");
#endif
#endif
}

__global__ __launch_bounds__(THREADS) void convcaps_votes_kernel(
    const float* __restrict__ poses,   // (4,16,16,32,4,4)
    const float* __restrict__ acts,    // (4,16,16,32,1,1)
    const float* __restrict__ kern,    // (3,3,32,32,4,4)
    float* __restrict__ out)           // votes flat ++ act_out flat
{
  __shared__ __align__(16) float lds_kern[SLAB];      // 64 KB: one (ky,kx) slab
  __shared__ __align__(16) float lds_pose[3 * ROWF];  // 18 KB: 3x3 pose patch

  const int t   = threadIdx.x;
  const int pix = blockIdx.x;                 // (b*OH + oh)*OW + ow
  const int ow  = pix % OWw;
  const int oh  = (pix / OWw) % OHh;
  const int b   = pix / (OWw * OHh);

  // poses strides (floats): mn=1, i=16, w=512, h=8192, b=131072.
  // Row ky of the 3x3 patch (kx=0..2, i, mn) is ROWF contiguous floats.
  const float* pose_base = poses + ((long)(b * H + oh) * W + ow) * (IC * ATOM);

  // ---- Stage the 3 pose-patch rows into LDS (async) ----
  for (int idx = t; idx < (3 * ROWF) / 4; idx += THREADS) {
    const int row = idx / (ROWF / 4);
    const int off = (idx % (ROWF / 4)) * 4;
    cp16_g2l(pose_base + (long)row * (W * IC * ATOM) + off,
             &lds_pose[row * ROWF + off]);
  }

  // ---- Activation patch output: 288 floats per pixel (tiny, L2 hits) ----
  if (t < NKK * IC) {
    const int kk = t >> 5, i = t & 31;
    const int ky = kk / 3, kx = kk % 3;
    const float a = acts[((long)(b * H + (oh + ky)) * W + (ow + kx)) * IC + i];
    __builtin_nontemporal_store(a, out + VOTES_TOTAL + (long)pix * (NKK * IC) + t);
  }

  float* vout_pix = out + (long)pix * PIXV;

  for (int kk = 0; kk < NKK; ++kk) {
    const int ky = kk / 3, kx = kk % 3;
    const float* ksrc = kern + (long)kk * SLAB;   // contiguous 64 KB slab

    __syncthreads();                              // lds_kern safe to overwrite
    #pragma unroll
    for (int c = 0; c < SLAB / 4 / THREADS; ++c) {   // 16 x b128 per thread
      const int off = (t + c * THREADS) * 4;
      cp16_g2l(ksrc + off, &lds_kern[off]);
    }
    async_wait_all();
    __syncthreads();                              // whole slab visible to WG

    if (kk + 1 < NKK) {                           // warm L2 for next slab
      __builtin_prefetch(ksrc + SLAB + t * 32, 0, 1);
      __builtin_prefetch(ksrc + SLAB + SLAB / 2 + t * 32, 0, 1);
    }

    // votes[pix][kk][f] = pose[ky,kx, i, mn] * kern[kk][f],  f = i*512+o*16+mn
    float* vout = vout_pix + kk * SLAB;
    const float* prow = &lds_pose[ky * ROWF + kx * (IC * ATOM)];
    #pragma unroll
    for (int c = 0; c < SLAB / 4 / THREADS; ++c) {
      const int f  = (t + c * THREADS) * 4;       // float4-aligned flat index
      const int i  = f >> 9;                      // / (OC*ATOM)
      const int mn = f & (ATOM - 1);              // in {0,4,8,12}
      const f4 pv = *(const f4*)&prow[i * ATOM + mn];   // broadcast over o
      const f4 kv = *(const f4*)&lds_kern[f];
      __builtin_nontemporal_store(pv * kv, (f4*)&vout[f]);  // streaming NT store
    }
  }
}

extern "C" void kernel_launch(void* const* d_in, const int* in_sizes, int n_in,
                              void* d_out, int out_size, void* d_ws, size_t ws_size,
                              hipStream_t stream) {
  const float* poses = (const float*)d_in[0];
  const float* acts  = (const float*)d_in[1];
  const float* kern  = (const float*)d_in[2];
  float* out         = (float*)d_out;
  convcaps_votes_kernel<<<NPIX, THREADS, 0, stream>>>(poses, acts, kern, out);
}